// CompressedInteractionNet_31997506355236
// MI455X (gfx1250) — compile-verified
//
#include <hip/hip_runtime.h>

typedef __attribute__((ext_vector_type(2))) float v2f;
typedef __attribute__((ext_vector_type(8))) float v8f;

// Problem constants (from setup_inputs): B=128, m=H=64, Hk=64, vk=16, D=16
#define BB    128
#define MM    64
#define DD    16
#define HK    64
#define VKK   16

__global__ __launch_bounds__(256) void cin_wmma_kernel(const float* __restrict__ x0,
                                                       const float* __restrict__ Vm,
                                                       const float* __restrict__ Vh,
                                                       float* __restrict__ out) {
    const int b    = blockIdx.x;          // one workgroup per batch
    const int wave = threadIdx.x >> 5;    // 8 waves
    const int lane = threadIdx.x & 31;
    const int n    = lane & 15;           // A-row (v) / B-column (d)
    const int hi   = lane >> 4;           // half: K-pair selector

    const float* x0b = x0 + b * (MM * DD);   // 1024 floats per batch

    // ---- Hoisted B-matrix fragments (x0-side), reused for all 8 heads ----
    // Transposed GEMMs:  a^T[v,d] = sum_i Vm^T[v,i] * x0r^T[i,d]
    //                    c^T[v,d] = sum_j Vh[v,j]   * X^T[j,d]
    // B layout (f32 4x16): VGPR0 = B[K=j0, N=n], VGPR1 = B[K=j0+1, N=n],
    //                      j0 = 4*kk + 2*hi
    v2f Bx0r[16];   // B-frag of x0r^T: B[i,d] = x_0[b,i,d]           (stride-64B pair)
    v2f BxT[16];    // B-frag of X^T:   B[j,d] = x0_flat[b, d*64+j]   (contiguous pair)
#pragma unroll
    for (int kk = 0; kk < 16; ++kk) {
        const int j0 = kk * 4 + hi * 2;
        Bx0r[kk].x = x0b[(j0 + 0) * DD + n];
        Bx0r[kk].y = x0b[(j0 + 1) * DD + n];
        const float* xr = x0b + n * MM + j0;
        BxT[kk].x = xr[0];
        BxT[kk].y = xr[1];
    }

    // ---- Each wave owns 8 consecutive heads k ----
    for (int kq = 0; kq < 8; ++kq) {
        const int k = wave * 8 + kq;
        const float* Vmk = Vm + k * (MM * VKK);     // [64 x 16] row-major (i, v)
        const float* Vhk = Vh + k * (VKK * MM);     // [16 x 64] row-major (v, j)

        v8f acc_a = {};   // a^T : rows v (VGPRs), cols d (lanes)
        v8f acc_c = {};   // c^T
#pragma unroll
        for (int kk = 0; kk < 16; ++kk) {
            const int j0 = kk * 4 + hi * 2;
            // A-frag of Vm^T: A[M=v, K=i] = Vm[i*16 + v]   (stride-64B pair)
            v2f am;
            am.x = Vmk[(j0 + 0) * VKK + n];
            am.y = Vmk[(j0 + 1) * VKK + n];
            // A-frag of Vh:   A[M=v, K=j] = Vh[v*64 + j]   (contiguous pair -> b64)
            v2f ah;
            const float* vr = Vhk + n * MM + j0;
            ah.x = vr[0];
            ah.y = vr[1];

            acc_a = __builtin_amdgcn_wmma_f32_16x16x4_f32(
                false, am, false, Bx0r[kk], (short)0, acc_a, false, false);
            acc_c = __builtin_amdgcn_wmma_f32_16x16x4_f32(
                false, ah, false, BxT[kk], (short)0, acc_c, false, false);
        }

        // out[b,k,d] = sum_v a^T[v,d]*c^T[v,d]
        // D layout: VGPR r -> row v=r (lanes 0-15) / v=r+8 (lanes 16-31), lane = col d
        float s = 0.0f;
#pragma unroll
        for (int r = 0; r < 8; ++r) s += acc_a[r] * acc_c[r];   // in-lane over 8 rows
        s += __shfl_xor(s, 16, 32);                             // fold the other half
        if (hi == 0)                                            // coalesced 16-lane store
            out[b * (HK * DD) + k * DD + n] = s;
    }
}

extern "C" void kernel_launch(void* const* d_in, const int* in_sizes, int n_in,
                              void* d_out, int out_size, void* d_ws, size_t ws_size,
                              hipStream_t stream) {
    (void)in_sizes; (void)n_in; (void)d_ws; (void)ws_size; (void)out_size;
    const float* x0 = (const float*)d_in[0];   // x_0 [128,64,16]
    // d_in[1] = x_h : unused by the reference (source bug recomputes it from x_0)
    const float* Vm = (const float*)d_in[2];   // [64,1,64,16]
    const float* Vh = (const float*)d_in[3];   // [64,1,16,64]
    float* out = (float*)d_out;                // [128,64,16]

    cin_wmma_kernel<<<BB, 256, 0, stream>>>(x0, Vm, Vh, out);
}